// CenterlineDiceLoss_31705448579090
// MI455X (gfx1250) — compile-verified
//
#include <hip/hip_runtime.h>
#include <math.h>

// -------- CDNA5 feature probes (compile-safe guards) -----------------------
#if __has_builtin(__builtin_amdgcn_global_load_async_to_lds_b32) && \
    __has_builtin(__builtin_amdgcn_s_wait_asynccnt)
#define HAVE_ASYNC_LDS 1
#else
#define HAVE_ASYNC_LDS 0
#endif

typedef __attribute__((ext_vector_type(2))) float v2f;
typedef __attribute__((ext_vector_type(8))) float v8f;

#define RAD 10
#define AXIS 160

__device__ __forceinline__ float fsigmoid(float x) {
    return 1.0f / (1.0f + __expf(-x));
}

// ---------------------------------------------------------------------------
// Pass 1: optional sigmoid + sliding-min (radius 10, clamped) along W.
// One block = one 160-element contiguous row, staged in LDS.
// Target pass uses async global->LDS loads (GLOBAL_LOAD_ASYNC_TO_LDS_B32,
// tracked by ASYNCcnt) when the builtin exists.
// ---------------------------------------------------------------------------
__global__ void __launch_bounds__(AXIS) minw_kernel(const float* __restrict__ in,
                                                    float* __restrict__ out,
                                                    int doSig) {
    __shared__ float row[AXIS];
    const int t = threadIdx.x;
    const int base = blockIdx.x * AXIS;

    if (doSig) {
        row[t] = fsigmoid(in[base + t]);
    } else {
#if HAVE_ASYNC_LDS
        __attribute__((address_space(1))) int* gp =
            (__attribute__((address_space(1))) int*)(in + base + t);
        __attribute__((address_space(3))) int* lp =
            (__attribute__((address_space(3))) int*)&row[t];
        __builtin_amdgcn_global_load_async_to_lds_b32(gp, lp, 0, 0);
        __builtin_amdgcn_s_wait_asynccnt(0);
#else
        row[t] = in[base + t];
#endif
    }
    __syncthreads();

    float m = row[t];
#pragma unroll
    for (int k = -RAD; k <= RAD; ++k) {
        int c = t + k;
        c = c < 0 ? 0 : (c > AXIS - 1 ? AXIS - 1 : c);
        m = fminf(m, row[c]);
    }
    out[base + t] = m;
}

// ---------------------------------------------------------------------------
// Pass 2: sliding-min along an axis with compile-time stride (here H: stride 160).
// Clamped-index reads; the 21x amplification is served by L2 (slab resident).
// ---------------------------------------------------------------------------
template <int STRIDE>
__global__ void __launch_bounds__(256) minaxis_kernel(const float* __restrict__ in,
                                                      float* __restrict__ out, int n) {
    const int i = blockIdx.x * 256 + threadIdx.x;
    if (i >= n) return;
    const int c = (i / STRIDE) % AXIS;
    float m = in[i];
#pragma unroll
    for (int k = -RAD; k <= RAD; ++k) {
        int cc = c + k;
        cc = cc < 0 ? 0 : (cc > AXIS - 1 ? AXIS - 1 : cc);
        m = fminf(m, in[i + (cc - c) * STRIDE]);
    }
    out[i] = m;
}

// ---------------------------------------------------------------------------
// Pass 3: sliding-min along D fused with the two reductions:
//   sProd = sum(skel * other)   sSkel = sum(skel)
// (other optionally passed through sigmoid on the fly). Deterministic block tree.
// ---------------------------------------------------------------------------
__global__ void __launch_bounds__(256) mind_reduce_kernel(const float* __restrict__ skelIn,
                                                          const float* __restrict__ other,
                                                          int otherSig,
                                                          float* __restrict__ pProd,
                                                          float* __restrict__ pSkel,
                                                          int n) {
    const int STRIDE = AXIS * AXIS;  // 25600
    const int tid = threadIdx.x;
    const int i = blockIdx.x * 256 + tid;

    float prod = 0.0f, sk = 0.0f;
    if (i < n) {
        const int c = (i / STRIDE) % AXIS;
        float m = skelIn[i];
#pragma unroll
        for (int k = -RAD; k <= RAD; ++k) {
            int cc = c + k;
            cc = cc < 0 ? 0 : (cc > AXIS - 1 ? AXIS - 1 : cc);
            m = fminf(m, skelIn[i + (cc - c) * STRIDE]);
        }
        float o = other[i];
        if (otherSig) o = fsigmoid(o);
        sk = m;
        prod = m * o;
    }

    __shared__ float s1[256];
    __shared__ float s2[256];
    s1[tid] = prod;
    s2[tid] = sk;
    __syncthreads();
    for (int off = 128; off > 0; off >>= 1) {
        if (tid < off) {
            s1[tid] += s1[tid + off];
            s2[tid] += s2[tid + off];
        }
        __syncthreads();
    }
    if (tid == 0) {
        pProd[blockIdx.x] = s1[0];
        pSkel[blockIdx.x] = s2[0];
    }
}

// ---------------------------------------------------------------------------
// Wave32 cross-lane sum via V_WMMA_F32_16X16X4_F32 with an all-ones B matrix:
// D[m,n] = rowsum_m(A); summing the 8 C/D VGPRs + SWAPX16 cross-half add gives
// the sum of all 64 A slots (= the 32 per-lane values; A vgpr1 is zero).
// EXEC is all-ones here (all 256 threads active, uniform trip counts).
// ---------------------------------------------------------------------------
__device__ __forceinline__ float wave_sum_wmma(float x) {
    v2f a;
    a[0] = x;
    a[1] = 0.0f;
    v2f b;
    b[0] = 1.0f;
    b[1] = 1.0f;
    v8f c = {};
    v8f d = __builtin_amdgcn_wmma_f32_16x16x4_f32(false, a, false, b, (short)0, c,
                                                  false, false);
    float r = d[0] + d[1] + d[2] + d[3] + d[4] + d[5] + d[6] + d[7];
    int rj = __builtin_amdgcn_ds_swizzle(__float_as_int(r), 0x401F);  // SWAPX16
    r += __int_as_float(rj);
    return r;
}

__global__ void __launch_bounds__(256) final_kernel(const float* __restrict__ pa,  // sum(pred_skel*target)
                                                    const float* __restrict__ pb,  // sum(target_skel)
                                                    const float* __restrict__ pc,  // sum(target_skel*p)
                                                    const float* __restrict__ pd,  // sum(pred_skel)
                                                    int nPart,
                                                    float* __restrict__ out) {
    const int tid = threadIdx.x;
    float s0 = 0.f, s1 = 0.f, s2 = 0.f, s3 = 0.f;
    for (int j = tid; j < nPart; j += 256) {  // nPart % 256 == 0 -> uniform EXEC
        s0 += pa[j];
        s1 += pb[j];
        s2 += pc[j];
        s3 += pd[j];
    }
    const float w0 = wave_sum_wmma(s0);
    const float w1 = wave_sum_wmma(s1);
    const float w2 = wave_sum_wmma(s2);
    const float w3 = wave_sum_wmma(s3);

    __shared__ float acc[8][4];
    const int wave = tid >> 5, lane = tid & 31;
    if (lane == 0) {
        acc[wave][0] = w0;
        acc[wave][1] = w1;
        acc[wave][2] = w2;
        acc[wave][3] = w3;
    }
    __syncthreads();
    if (tid == 0) {
        float SA = 0.f, SB = 0.f, SC = 0.f, SD = 0.f;
        for (int w = 0; w < 8; ++w) {
            SA += acc[w][0];
            SB += acc[w][1];
            SC += acc[w][2];
            SD += acc[w][3];
        }
        const float tprec = (SA + 1.0f) / (SB + 1.0f);
        const float tsens = (SC + 1.0f) / (SD + 1.0f);
        const float cld = 2.0f * tprec * tsens / (tprec + tsens + 1.0f);
        out[0] = 1.0f - cld;
    }
}

// ---------------------------------------------------------------------------
extern "C" void kernel_launch(void* const* d_in, const int* in_sizes, int n_in,
                              void* d_out, int out_size, void* d_ws, size_t ws_size,
                              hipStream_t stream) {
    (void)n_in;
    (void)out_size;
    (void)ws_size;

    const float* pred = (const float*)d_in[0];
    const float* target = (const float*)d_in[1];
    float* out = (float*)d_out;

    const int N = in_sizes[0];  // 4*1*160*160*160 = 16,384,000
    float* WS1 = (float*)d_ws;
    float* WS2 = WS1 + N;
    const int nRows = N / AXIS;           // 102,400 rows
    const int nBlk = (N + 255) / 256;     // 64,000 blocks (exact)
    float* pProdPred = WS2 + N;           // sum(pred_skel * target)
    float* pSkelPred = pProdPred + nBlk;  // sum(pred_skel)
    float* pProdTgt = pSkelPred + nBlk;   // sum(target_skel * p)
    float* pSkelTgt = pProdTgt + nBlk;    // sum(target_skel)

    // pred pipeline: sigmoid fused into W pass; skel never materialized
    minw_kernel<<<nRows, AXIS, 0, stream>>>(pred, WS1, 1);
    minaxis_kernel<AXIS><<<nBlk, 256, 0, stream>>>(WS1, WS2, N);
    mind_reduce_kernel<<<nBlk, 256, 0, stream>>>(WS2, target, 0, pProdPred, pSkelPred, N);

    // target pipeline: sigmoid(pred) recomputed on the fly in the fused D pass
    minw_kernel<<<nRows, AXIS, 0, stream>>>(target, WS1, 0);
    minaxis_kernel<AXIS><<<nBlk, 256, 0, stream>>>(WS1, WS2, N);
    mind_reduce_kernel<<<nBlk, 256, 0, stream>>>(WS2, pred, 1, pProdTgt, pSkelTgt, N);

    final_kernel<<<1, 256, 0, stream>>>(pProdPred, pSkelTgt, pProdTgt, pSkelPred, nBlk, out);
}